// LogTransform_75711683494100
// MI455X (gfx1250) — compile-verified
//
#include <hip/hip_runtime.h>
#include <math.h>
#include <stdint.h>

// ---------------------------------------------------------------------------
// logm(X) for SPD X via inverse scaling-and-squaring + Newton-Schulz sqrt.
// All heavy lifting: batched f32 WMMA GEMM (V_WMMA_F32_16X16X4_F32), LDS
// double-buffered, tiles filled with GLOBAL_LOAD_ASYNC_TO_LDS_B128 (ASYNCcnt).
// Every B-side matrix in this algorithm is symmetric (polynomials of the SPD
// input), so the WMMA B-operand (columns of B) is read from B's rows directly:
// no transpose anywhere, both tiles are straight K-contiguous b128 copies.
// ---------------------------------------------------------------------------

typedef __attribute__((ext_vector_type(2))) float v2f;
typedef __attribute__((ext_vector_type(8))) float v8f;
typedef __attribute__((__vector_size__(4 * sizeof(int)))) int v4i_;

#define NMAT 512
#define NN   (NMAT * NMAT)
#define TILE 128
#define KT   32
#define LDT  36    // LDS tile row stride (floats): 144B rows keep b128 align;
                   // gcd(36,64)=4 -> 16-bank walk, conflict-free v2f fragments.
#define NTILES (NMAT / KT)

#if defined(__gfx1250__) &&                                                    \
    __has_builtin(__builtin_amdgcn_global_load_async_to_lds_b128) &&           \
    __has_builtin(__builtin_amdgcn_s_wait_asynccnt)
#define USE_ASYNC_LDS 1
#else
#define USE_ASYNC_LDS 0
#endif

#if USE_ASYNC_LDS
__device__ __forceinline__ void async_cp16(const float* g, float* l) {
  __attribute__((address_space(1))) v4i_* gp =
      (__attribute__((address_space(1))) v4i_*)const_cast<float*>(g);
  __attribute__((address_space(3))) v4i_* lp =
      (__attribute__((address_space(3))) v4i_*)l;
  __builtin_amdgcn_global_load_async_to_lds_b128(gp, lp, 0, 0);
}
#endif

// Copy a 128x32 tile (rows row0..row0+127, cols kb..kb+31 of a 512x512
// row-major matrix) into LDS [row][K] with stride LDT. 4 b128 ops per thread.
__device__ __forceinline__ void fill_half(const float* __restrict__ G,
                                          size_t boff, int row0, int kb,
                                          float* Ls, int t) {
#pragma unroll
  for (int i = 0; i < 4; ++i) {           // 1024 float4 slots / 256 threads
    int s   = t + i * 256;
    int row = s >> 3;
    int c4  = (s & 7) << 2;
    const float* g = G + boff + (size_t)(row0 + row) * NMAT + kb + c4;
    float* l = &Ls[row * LDT + c4];
#if USE_ASYNC_LDS
    async_cp16(g, l);
#else
    *(float4*)l = *(const float4*)g;
#endif
  }
}

// ---------------------------------------------------------------------------
// Batched GEMM (B symmetric): D = alpha * A*B + delta * I
// (512x512 f32, batch = blockIdx.z). 256 threads = 8 waves; 128x128 workgroup
// tile; each wave owns 32x64 = 2x4 WMMA 16x16 f32 accumulators.
// ---------------------------------------------------------------------------
__global__ __launch_bounds__(256) void k_gemm(const float* __restrict__ A,
                                              const float* __restrict__ Bm,
                                              float* __restrict__ D,
                                              float alpha, float delta) {
  __shared__ __align__(16) float As[2][TILE * LDT];  // A rows   [M][K]
  __shared__ __align__(16) float Bs[2][TILE * LDT];  // B rows = B cols (sym)

  const size_t boff = (size_t)blockIdx.z * NN;
  const int tileM = blockIdx.x * TILE;
  const int tileN = blockIdx.y * TILE;
  const int t    = threadIdx.x;
  const int wid  = t >> 5;
  const int lane = t & 31;
  const int half = lane >> 4;   // 0: lanes 0-15, 1: lanes 16-31
  const int l16  = lane & 15;
  const int waveM = (wid & 3) * 32;   // 4 waves along M
  const int waveN = (wid >> 2) * 64;  // 2 waves along N

  v8f acc[2][4];
#pragma unroll
  for (int mi = 0; mi < 2; ++mi)
#pragma unroll
    for (int ni = 0; ni < 4; ++ni)
      acc[mi][ni] = (v8f){};

  fill_half(A, boff, tileM, 0, As[0], t);
  fill_half(Bm, boff, tileN, 0, Bs[0], t);

  for (int ti = 0; ti < NTILES; ++ti) {
#if USE_ASYNC_LDS
    __builtin_amdgcn_s_wait_asynccnt(0);   // own async LDS writes complete
#endif
    __syncthreads();                       // publish tile ti to all waves
    if (ti + 1 < NTILES) {                 // prefetch overlaps WMMA below
      fill_half(A, boff, tileM, (ti + 1) * KT, As[(ti + 1) & 1], t);
      fill_half(Bm, boff, tileN, (ti + 1) * KT, Bs[(ti + 1) & 1], t);
    }

    const float* Ab = As[ti & 1];
    const float* Bb = Bs[ti & 1];
#pragma unroll
    for (int k0 = 0; k0 < KT; k0 += 4) {
      // f32 WMMA fragments (ISA 7.12.2): lane<16 holds {K=k0,k0+1},
      // lane>=16 holds {K=k0+2,k0+3} of its row(A)/col(B, via symmetry).
      const int ka = k0 + 2 * half;
      v2f afr[2], bfr[4];
#pragma unroll
      for (int mi = 0; mi < 2; ++mi)
        afr[mi] = *(const v2f*)&Ab[(waveM + mi * 16 + l16) * LDT + ka];
#pragma unroll
      for (int ni = 0; ni < 4; ++ni)
        bfr[ni] = *(const v2f*)&Bb[(waveN + ni * 16 + l16) * LDT + ka];
#pragma unroll
      for (int mi = 0; mi < 2; ++mi)
#pragma unroll
        for (int ni = 0; ni < 4; ++ni)
          acc[mi][ni] = __builtin_amdgcn_wmma_f32_16x16x4_f32(
              false, afr[mi], false, bfr[ni], (short)0, acc[mi][ni], false, false);
    }
    __syncthreads();  // done reading buf[ti&1]; next prefetch may overwrite it
  }

  // Epilogue: C/D layout — VGPR r: lanes0-15 -> (M=r,N=lane), lanes16-31 -> M=r+8
#pragma unroll
  for (int mi = 0; mi < 2; ++mi)
#pragma unroll
    for (int ni = 0; ni < 4; ++ni)
#pragma unroll
      for (int r = 0; r < 8; ++r) {
        int row = tileM + waveM + mi * 16 + r + 8 * half;
        int col = tileN + waveN + ni * 16 + l16;
        float v = alpha * acc[mi][ni][r];
        if (row == col) v += delta;
        D[boff + (size_t)row * NMAT + col] = v;
      }
}

// ---------------------------------------------------------------------------
// Helpers
// ---------------------------------------------------------------------------
__global__ __launch_bounds__(256) void k_trace(const float* __restrict__ x,
                                               float* __restrict__ tbuf,
                                               float* __restrict__ logt) {
  __shared__ float red[256];
  const int b = blockIdx.x;
  const float* xb = x + (size_t)b * NN;
  float s = 0.f;
  for (int i = threadIdx.x; i < NMAT; i += 256) s += xb[(size_t)i * NMAT + i];
  red[threadIdx.x] = s;
  __syncthreads();
  for (int off = 128; off > 0; off >>= 1) {
    if (threadIdx.x < off) red[threadIdx.x] += red[threadIdx.x + off];
    __syncthreads();
  }
  if (threadIdx.x == 0) {
    float tv = red[0];
    if (!(tv > 1e-30f)) tv = 1e-30f;
    tbuf[b] = tv;
    logt[b] = logf(tv);
  }
}

__global__ __launch_bounds__(256) void k_scale(const float* __restrict__ x,
                                               const float* __restrict__ tbuf,
                                               float* __restrict__ Y,
                                               int b0, int n) {
  int idx = blockIdx.x * 256 + threadIdx.x;
  if (idx >= n) return;
  int b = idx / NN;
  Y[idx] = x[idx] * (1.0f / tbuf[b0 + b]);
}

__global__ __launch_bounds__(256) void k_setid(float* __restrict__ P, float c, int n) {
  int idx = blockIdx.x * 256 + threadIdx.x;
  if (idx >= n) return;
  int l = idx % NN;
  P[idx] = ((l / NMAT) == (l % NMAT)) ? c : 0.0f;
}

__global__ __launch_bounds__(256) void k_subI(const float* __restrict__ M,
                                              float* __restrict__ E, int n) {
  int idx = blockIdx.x * 256 + threadIdx.x;
  if (idx >= n) return;
  int l = idx % NN;
  E[idx] = M[idx] - (((l / NMAT) == (l % NMAT)) ? 1.0f : 0.0f);
}

__global__ __launch_bounds__(256) void k_diagadd(float* __restrict__ outc,
                                                 const float* __restrict__ logt,
                                                 int b0, int n) {
  int idx = blockIdx.x * 256 + threadIdx.x;  // n = bc * NMAT
  if (idx >= n) return;
  int b = idx / NMAT, d = idx % NMAT;
  outc[(size_t)b * NN + (size_t)d * NMAT + d] += logt[b0 + b];
}

// ---------------------------------------------------------------------------
extern "C" void kernel_launch(void* const* d_in, const int* in_sizes, int n_in,
                              void* d_out, int out_size, void* d_ws, size_t ws_size,
                              hipStream_t stream) {
  const float* x = (const float*)d_in[0];
  float* out = (float*)d_out;
  const int Btot = in_sizes[0] / NN;

  // Workspace layout: [tbuf|logt (1KB head)] + 5 chunk-sized N*N*bchunk buffers
  uint8_t* base = (uint8_t*)d_ws;
  float* tbuf = (float*)base;
  float* logt = tbuf + Btot;
  const size_t head = 1024;
  const size_t matBytes = (size_t)NN * sizeof(float);
  size_t avail = (ws_size > head) ? (ws_size - head) : 0;
  int bchunk = (int)(avail / (5 * matBytes));
  if (bchunk < 1) bchunk = 1;
  if (bchunk > Btot) bchunk = Btot;

  float* buf[5];
  for (int i = 0; i < 5; ++i)
    buf[i] = (float*)(base + head + (size_t)i * (size_t)bchunk * matBytes);

  k_trace<<<Btot, 256, 0, stream>>>(x, tbuf, logt);

  const int KSQ = 5;   // sqrt stages: M = A0^(1/32)
  const int NSI = 12;  // Newton-Schulz iterations per sqrt
  const int SM  = 8;   // log-series terms

  for (int b0 = 0; b0 < Btot; b0 += bchunk) {
    const int bc = (Btot - b0 < bchunk) ? (Btot - b0) : bchunk;
    const int ne = bc * NN;
    const int eb = (ne + 255) / 256;
    dim3 gg(NMAT / TILE, NMAT / TILE, bc);

    float *Yc = buf[0], *Ya = buf[1], *Zc = buf[2], *Za = buf[3], *T = buf[4];

    // A0 = X / trace(X)
    k_scale<<<eb, 256, 0, stream>>>(x + (size_t)b0 * NN, tbuf, Yc, b0, ne);

    // Repeated sqrt via coupled Newton-Schulz: Y->sqrt(M), Z->inv sqrt(M)
    // (all iterates are symmetric polynomials of A0)
    for (int s = 0; s < KSQ; ++s) {
      k_setid<<<eb, 256, 0, stream>>>(Zc, 1.0f, ne);
      for (int it = 0; it < NSI; ++it) {
        k_gemm<<<gg, 256, 0, stream>>>(Zc, Yc, T, -0.5f, 1.5f);  // T = 1.5I - .5 Z*Y
        k_gemm<<<gg, 256, 0, stream>>>(Yc, T, Ya, 1.0f, 0.0f);   // Y' = Y*T
        k_gemm<<<gg, 256, 0, stream>>>(T, Zc, Za, 1.0f, 0.0f);   // Z' = T*Z
        float* tp;
        tp = Yc; Yc = Ya; Ya = tp;
        tp = Zc; Zc = Za; Za = tp;
      }
    }

    // E = M - I ;  log(M) = E*(c1 I + E*(c2 I + ... )) via Horner
    float* E = Za;
    k_subI<<<eb, 256, 0, stream>>>(Yc, E, ne);
    float* Pc = T;
    float* Pa = Zc;
    float cm = (((SM & 1) ? 1.0f : -1.0f)) / (float)SM;
    k_setid<<<eb, 256, 0, stream>>>(Pc, cm, ne);
    for (int j = SM - 1; j >= 1; --j) {
      float cj = (((j & 1) ? 1.0f : -1.0f)) / (float)j;
      k_gemm<<<gg, 256, 0, stream>>>(E, Pc, Pa, 1.0f, cj);  // P = E*P + cj I
      float* tp = Pc; Pc = Pa; Pa = tp;
    }

    // out = 2^k * E*P  (then add log(trace) on the diagonal)
    const float s2k = (float)(1 << KSQ);
    k_gemm<<<gg, 256, 0, stream>>>(E, Pc, out + (size_t)b0 * NN, s2k, 0.0f);
    const int dn = bc * NMAT;
    k_diagadd<<<(dn + 255) / 256, 256, 0, stream>>>(out + (size_t)b0 * NN,
                                                    logt, b0, dn);
  }
}